// SkeletonExtractionLayer_62835371540925
// MI455X (gfx1250) — compile-verified
//
#include <hip/hip_runtime.h>

// Zhang-Suen skeletonization, bit-packed, fully resident in LDS.
// One workgroup per image (32, 512, 512): image = 8192 x 32-bit words (32 KB),
// stored with a 1-word halo ring so out-of-range neighbors read as 0
// (matches the reference's zero padding) with no bounds checks.

namespace {
constexpr int ROWS = 512;
constexpr int COLS = 512;
constexpr int WPR  = COLS / 32;          // 16 words per row
constexpr int PROW = WPR + 2;            // 18 (halo word on each side)
constexpr int PBUF = (ROWS + 2) * PROW;  // 9252 words = 37008 B LDS
constexpr int NWORDS = ROWS * WPR;       // 8192 interior words
constexpr int NTH = 1024;                // 32 wave32 waves
constexpr int WPT = NWORDS / NTH;        // 8 words per thread per substep
constexpr int SUBSTEPS = 16;             // ITERATIONS(8) * 2
}

__global__ __launch_bounds__(NTH, 1)
void skeleton_bitpar_kernel(const float* __restrict__ in, float* __restrict__ out) {
    __shared__ unsigned buf[PBUF];

    const int tid  = threadIdx.x;
    const int lane = tid & 31;
    const int wave = tid >> 5;
    const size_t ibase = (size_t)blockIdx.x * (size_t)(ROWS * COLS);

    // ---- 1) zero LDS (halo ring must stay 0 = reference zero padding) ----
    for (int i = tid; i < PBUF; i += NTH) buf[i] = 0u;
    __syncthreads();

    // ---- 2) threshold + bit-pack: 32 lanes -> one word via wave32 ballot ----
    constexpr int ROUNDS = NWORDS / 32;  // 256 words per wave
    for (int k = 0; k < ROUNDS; ++k) {
        const int wid = wave * ROUNDS + k;
        const int row = wid >> 4;
        const int wc  = wid & (WPR - 1);
        const float* p = in + ibase + (size_t)row * COLS + wc * 32 + lane;
        __builtin_prefetch(p + 4 * COLS, 0, 0);  // global_prefetch_b8 (speculative)
        const unsigned w = __builtin_amdgcn_ballot_w32(p[0] >= 0.5f);
        if (lane == 0) buf[(row + 1) * PROW + (wc + 1)] = w;
    }
    __syncthreads();

    // ---- 3) 16 bit-parallel thinning substeps, all in LDS ----
    for (int s = 0; s < SUBSTEPS; ++s) {
        const bool first = (s & 1) == 0;
        unsigned neww[WPT];

        #pragma unroll
        for (int k = 0; k < WPT; ++k) {
            const int wid = k * NTH + tid;
            const int row = wid >> 4;
            const int wc  = wid & (WPR - 1);
            const int ci  = (row + 1) * PROW + (wc + 1);

            const unsigned aL = buf[ci - PROW - 1], aM = buf[ci - PROW], aR = buf[ci - PROW + 1];
            const unsigned cL = buf[ci - 1],        cM = buf[ci],        cR = buf[ci + 1];
            const unsigned bL = buf[ci + PROW - 1], bM = buf[ci + PROW], bR = buf[ci + PROW + 1];

            // bit i of word = pixel column wc*32 + i; west = bit i-1 (carry from left word)
            const unsigned Nv  = aM;
            const unsigned Sv  = bM;
            const unsigned Wv  = (cM << 1) | (cL >> 31);
            const unsigned Ev  = (cM >> 1) | (cR << 31);
            const unsigned NWv = (aM << 1) | (aL >> 31);
            const unsigned NEv = (aM >> 1) | (aR << 31);
            const unsigned SWv = (bM << 1) | (bL >> 31);
            const unsigned SEv = (bM >> 1) | (bR << 31);

            // clockwise neighbor ring P2..P9 = N, NE, E, SE, S, SW, W, NW
            unsigned n[8] = {Nv, NEv, Ev, SEv, Sv, SWv, Wv, NWv};

            // bcnt >= 2  (at least two neighbor bits set)
            unsigned acc = n[0], ge2 = 0u;
            #pragma unroll
            for (int i = 1; i < 8; ++i) { ge2 |= acc & n[i]; acc |= n[i]; }

            // bcnt <= 6  <=>  at least two neighbor bits CLEAR
            unsigned accz = ~n[0], le6 = 0u;
            #pragma unroll
            for (int i = 1; i < 8; ++i) { const unsigned z = ~n[i]; le6 |= accz & z; accz |= z; }

            // acnt == 1: exactly one 0->1 transition around the ring
            unsigned t[8];
            #pragma unroll
            for (int i = 0; i < 8; ++i) t[i] = ~n[i] & n[(i + 1) & 7];
            unsigned ta = t[0], tg2 = 0u;
            #pragma unroll
            for (int i = 1; i < 8; ++i) { tg2 |= ta & t[i]; ta |= t[i]; }
            const unsigned ex1 = ta & ~tg2;

            // substep-specific conditions (P2=n0, P4=n2, P6=n4, P8=n6)
            const unsigned c1 = first ? ~(n[0] & n[2] & n[4]) : ~(n[0] & n[2] & n[6]);
            const unsigned c2 = first ? ~(n[2] & n[4] & n[6]) : ~(n[0] & n[4] & n[6]);

            neww[k] = cM & ~(ge2 & le6 & ex1 & c1 & c2);
        }

        __syncthreads();  // all reads of this substep complete

        #pragma unroll
        for (int k = 0; k < WPT; ++k) {
            const int wid = k * NTH + tid;
            const int row = wid >> 4;
            const int wc  = wid & (WPR - 1);
            buf[(row + 1) * PROW + (wc + 1)] = neww[k];
        }

        __syncthreads();  // all writes visible before next substep
    }

    // ---- 4) unpack bits -> 0.0/1.0 floats, coalesced 128B stores per wave ----
    for (int k = 0; k < ROUNDS; ++k) {
        const int wid = wave * ROUNDS + k;
        const int row = wid >> 4;
        const int wc  = wid & (WPR - 1);
        const unsigned w = buf[(row + 1) * PROW + (wc + 1)];  // LDS broadcast
        out[ibase + (size_t)row * COLS + wc * 32 + lane] =
            ((w >> lane) & 1u) ? 1.0f : 0.0f;
    }
}

extern "C" void kernel_launch(void* const* d_in, const int* in_sizes, int n_in,
                              void* d_out, int out_size, void* d_ws, size_t ws_size,
                              hipStream_t stream) {
    (void)n_in; (void)out_size; (void)d_ws; (void)ws_size;
    const float* in = (const float*)d_in[0];
    float* out = (float*)d_out;
    const int nimg = in_sizes[0] / (ROWS * COLS);  // 32
    skeleton_bitpar_kernel<<<nimg, NTH, 0, stream>>>(in, out);
}